// DecoderWrapper_58317065945251
// MI455X (gfx1250) — compile-verified
//
#include <hip/hip_runtime.h>
#include <stdint.h>

typedef float v2f __attribute__((ext_vector_type(2)));
typedef float v8f __attribute__((ext_vector_type(8)));

// Exact-fp32 CDNA5 matrix op: D(16x16,f32) = A(16x4,f32) * B(4x16,f32) + C
static __device__ inline v8f wmma4(v2f a, v2f b, v8f c) {
  return __builtin_amdgcn_wmma_f32_16x16x4_f32(false, a, false, b, (short)0, c,
                                               false, false);
}

#define CDIM 256      // embed dim
#define NEMB 16384    // codebook size
#define EROW 260      // padded LDS row stride (floats): conflict-free ds_load_b64

// ---------------------------------------------------------------------------
// ||e||^2 per codebook row
// ---------------------------------------------------------------------------
__global__ void enorm_kernel(const float* __restrict__ embed,
                             float* __restrict__ enorm) {
  int n = blockIdx.x * blockDim.x + threadIdx.x;
  if (n >= NEMB) return;
  const float* row = embed + (size_t)n * CDIM;
  float s = 0.f;
  for (int c = 0; c < CDIM; ++c) { float v = row[c]; s += v * v; }
  enorm[n] = s;
}

// ---------------------------------------------------------------------------
// Weight repacks into WMMA-fragment-friendly layout:
//   Bw[((k>>2)*Nout + n)*4 + (k&3)] = Wt[k][n]
// ---------------------------------------------------------------------------
__global__ void repack_pq_kernel(const float* __restrict__ w,  // [256][256] (OI)
                                 float* __restrict__ Bw) {
  int i = blockIdx.x * 256 + threadIdx.x;
  if (i >= CDIM * CDIM) return;
  int k = i >> 8, n = i & 255;
  Bw[(((k >> 2) * CDIM + n) << 2) + (k & 3)] = w[n * CDIM + k];
}

__global__ void repack_conv3_kernel(const float* __restrict__ w,  // OIHW 3x3
                                    float* __restrict__ Bw, int Cin, int Nout) {
  int total = 9 * Cin * Nout;
  int i = blockIdx.x * 256 + threadIdx.x;
  if (i >= total) return;
  int k = i / Nout, n = i % Nout;
  int tap = k / Cin, c = k % Cin;             // k = tap*Cin + c
  Bw[(((k >> 2) * Nout + n) << 2) + (k & 3)] = w[(n * Cin + c) * 9 + tap];
}

__global__ void repack_convout_kernel(const float* __restrict__ w,  // [3][64][3][3]
                                      const float* __restrict__ bias3,
                                      float* __restrict__ Bw,
                                      float* __restrict__ bpad) {
  const int Cin = 64, Np = 16, total = 9 * Cin * Np;
  int i = blockIdx.x * 256 + threadIdx.x;
  if (i < Np) bpad[i] = (i < 3) ? bias3[i] : 0.f;
  if (i >= total) return;
  int k = i / Np, n = i % Np;
  int tap = k / Cin, c = k % Cin;
  Bw[(((k >> 2) * Np + n) << 2) + (k & 3)] =
      (n < 3) ? w[(n * Cin + c) * 9 + tap] : 0.f;
}

// ---------------------------------------------------------------------------
// VQ: fused distance GEMM + argmin.  One block = 16 rows of z_flat, 4 waves.
// A panel (16x256) resident in 64 v2f fragments, reused across 256 N tiles.
// Embed panels stream through per-wave double-buffered LDS tiles filled with
// global_load_async_to_lds_b128 (ASYNCcnt), so global latency is overlapped
// with a full tile of WMMA work and the hot loop only waits on DS.
// ---------------------------------------------------------------------------
__global__ __launch_bounds__(128, 1) void vq_argmin_kernel(
    const float* __restrict__ z, const float* __restrict__ embed,
    const float* __restrict__ enorm, int* __restrict__ idx_out) {
  __shared__ float zt[16][CDIM];
  __shared__ float ebuf[4][2][16 * EROW];   // per-wave double buffer
  __shared__ float rmin[4][16];
  __shared__ int ridx[4][16];
  const int tid = threadIdx.x;
  const int m0 = blockIdx.x * 16;

  // stage z tile: z is NCHW (4,256,32,32); row m -> (b, h*32+w)
  for (int i = tid; i < 16 * CDIM; i += 128) {
    int r = i >> 8, c = i & (CDIM - 1);
    int m = m0 + r;
    int b = m >> 10;
    int hw = m & 1023;
    zt[r][c] = z[(size_t)b * (CDIM * 1024) + (size_t)c * 1024 + hw];
  }
  __syncthreads();

  const int lane = tid & 31, wave = tid >> 5;
  const int half = lane >> 4, lr = lane & 15;

  v2f A[64];
#pragma unroll
  for (int kk = 0; kk < 64; ++kk) {
    A[kk].x = zt[lr][kk * 4 + 2 * half];
    A[kk].y = zt[lr][kk * 4 + 2 * half + 1];
  }

  float minv[8];
  int mini[8];
#pragma unroll
  for (int r = 0; r < 8; ++r) { minv[r] = 3.4e38f; mini[r] = 0x7fffffff; }

  const uint32_t lbase0 = (uint32_t)(uintptr_t)&ebuf[wave][0][0];
  const uint32_t lbase1 = (uint32_t)(uintptr_t)&ebuf[wave][1][0];
  const uint64_t ebytes = (uint64_t)(uintptr_t)embed;
  const int nbeg = wave * (NEMB / 4);          // 4096 codes / wave, 256 tiles

  // Issue one 16-row embed panel (16KB) as 32 async b128 copies (512B/wave ea).
  auto issue_tile = [&](int n0, uint32_t lb) {
    uint64_t gb = ebytes + (uint64_t)n0 * (CDIM * 4) + (uint32_t)(lane * 16);
    uint32_t lo = lb + (uint32_t)(lane * 16);
#pragma unroll
    for (int j = 0; j < 32; ++j) {
      uint64_t g = gb + (uint32_t)((j >> 1) * (CDIM * 4) + (j & 1) * 512);
      uint32_t l = lo + (uint32_t)((j >> 1) * (EROW * 4) + (j & 1) * 512);
      asm volatile("global_load_async_to_lds_b128 %0, %1, off"
                   :: "v"(l), "v"(g) : "memory");
    }
  };

  issue_tile(nbeg, lbase0);
  for (int t = 0; t < NEMB / 4 / 16; ++t) {
    const int n0 = nbeg + t * 16;
    const int cur = t & 1;
    if (t < NEMB / 4 / 16 - 1) {
      issue_tile(n0 + 16, cur ? lbase0 : lbase1);
      asm volatile("s_wait_asynccnt 0x20" ::: "memory");  // tile t complete
    } else {
      asm volatile("s_wait_asynccnt 0x0" ::: "memory");
    }
    const float* bbase = &ebuf[wave][cur][lr * EROW + 2 * half];
    v8f acc = {};
#pragma unroll
    for (int g = 0; g < 8; ++g) {
      v2f bf[8];
#pragma unroll
      for (int j = 0; j < 8; ++j)
        bf[j] = *(const v2f*)(bbase + (g * 8 + j) * 4);
#pragma unroll
      for (int j = 0; j < 8; ++j) acc = wmma4(A[g * 8 + j], bf[j], acc);
    }
    float en = enorm[n0 + lr];
    int cand = n0 + lr;
#pragma unroll
    for (int r = 0; r < 8; ++r) {
      float s = en - 2.0f * acc[r];   // ||z||^2 omitted: constant per row
      if (s < minv[r]) { minv[r] = s; mini[r] = cand; }
    }
  }

  // reduce across the 16 lanes of each half (row r lives in lanes 0-15,
  // row 8+r in lanes 16-31); masks 1,2,4,8 stay inside each half.
#pragma unroll
  for (int msk = 1; msk < 16; msk <<= 1) {
#pragma unroll
    for (int r = 0; r < 8; ++r) {
      float ov = __shfl_xor(minv[r], msk, 32);
      int oi = __shfl_xor(mini[r], msk, 32);
      if (ov < minv[r] || (ov == minv[r] && oi < mini[r])) {
        minv[r] = ov; mini[r] = oi;
      }
    }
  }
  if (lr == 0) {
#pragma unroll
    for (int r = 0; r < 8; ++r) {
      rmin[wave][r + 8 * half] = minv[r];
      ridx[wave][r + 8 * half] = mini[r];
    }
  }
  __syncthreads();
  if (tid < 16) {
    float bv = rmin[0][tid];
    int bi = ridx[0][tid];
    for (int w2 = 1; w2 < 4; ++w2) {
      float v = rmin[w2][tid]; int i2 = ridx[w2][tid];
      if (v < bv || (v == bv && i2 < bi)) { bv = v; bi = i2; }
    }
    idx_out[m0 + tid] = bi;
  }
}

// ---------------------------------------------------------------------------
// pq 1x1 conv as gather-GEMM: A[m][k] = embed[idx[m]][k], z_q never built.
// ---------------------------------------------------------------------------
__global__ __launch_bounds__(128) void pq_gemm_kernel(
    const float* __restrict__ embed, const int* __restrict__ idx,
    const float* __restrict__ Bw, const float* __restrict__ bias,
    float* __restrict__ q) {
  int wid = blockIdx.x * 4 + (threadIdx.x >> 5);
  int mt = wid >> 4, nt = wid & 15;
  int lane = threadIdx.x & 31, half = lane >> 4, lr = lane & 15;
  int m0 = mt * 16, n0 = nt * 16;

  const float* arow = embed + (size_t)idx[m0 + lr] * CDIM + 2 * half;
  const float* bbase = Bw + (size_t)(n0 + lr) * 4 + 2 * half;
  v8f acc = {};
#pragma unroll
  for (int g = 0; g < 8; ++g) {       // 8 groups of 8 K-chunks, batched loads
    v2f a[8], bf[8];
#pragma unroll
    for (int j = 0; j < 8; ++j) {
      int kk = g * 8 + j;
      a[j] = *(const v2f*)(arow + kk * 4);
      bf[j] = *(const v2f*)(bbase + (size_t)kk * CDIM * 4);
    }
#pragma unroll
    for (int j = 0; j < 8; ++j) acc = wmma4(a[j], bf[j], acc);
  }
  float bv = bias[n0 + lr];
#pragma unroll
  for (int r = 0; r < 8; ++r) {
    int mm = m0 + r + 8 * half;
    q[(size_t)mm * CDIM + n0 + lr] = acc[r] + bv;
  }
}

// ---------------------------------------------------------------------------
// Big 3x3 convs: block = one 16-pixel M-tile x 4 N-tiles (one per wave).
// Per tap, the shared 16xCin A panel is staged cooperatively into a
// double-buffered LDS tile via global_load_async_to_lds_b128; taps pipeline
// (issue tap t+1, compute tap t, s_wait_asynccnt + barrier).  A fragments
// come from conflict-free padded LDS rows; B weight fragments are batched
// L2-hot global loads.  Out-of-image rows are clamped at stage time and
// zeroed at read time per-lane (select, EXEC stays full for WMMA).
// ---------------------------------------------------------------------------
template <int CIN, bool UPSAMPLE, bool RELU>
__global__ __launch_bounds__(128, 1) void conv3s_kernel(
    const float* __restrict__ in, const float* __restrict__ Bw,
    const float* __restrict__ bias, float* __restrict__ out, int Ho, int Wo,
    int Nout) {
  constexpr int AROW = CIN + 4;          // padded row stride (floats)
  __shared__ float abuf[2][16 * AROW];
  const int tid = threadIdx.x;
  const int lane = tid & 31, wave = tid >> 5;
  const int half = lane >> 4, lr = lane & 15;

  const int ngb = (Nout >> 4) >> 2;      // N-tile groups per M-tile (>=1)
  const int mt = blockIdx.x / ngb;
  const int ntg = blockIdx.x % ngb;
  const int m0 = mt * 16;
  const int n0 = (ntg * 4 + wave) * 16;

  const int hw = Ho * Wo;
  const int Hi = UPSAMPLE ? (Ho >> 1) : Ho;
  const int Wi = UPSAMPLE ? (Wo >> 1) : Wo;

  // staging coords: thread handles row r = tid>>3, segment seg = tid&7
  const int r_s = tid >> 3, seg = tid & 7;
  const int mm_s = m0 + r_s;
  const int b_s = mm_s / hw;
  const int rem_s = mm_s % hw;
  const int y_s = rem_s / Wo, x_s = rem_s % Wo;

  // compute coords: lane owns A row lr (its output pixel)
  const int mm_l = m0 + lr;
  const int b_l = mm_l / hw;
  const int rem_l = mm_l % hw;
  const int y_l = rem_l / Wo, x_l = rem_l % Wo;

  const uint64_t inB = (uint64_t)(uintptr_t)in;
  const uint32_t lb0 = (uint32_t)(uintptr_t)&abuf[0][0];
  const uint32_t lb1 = (uint32_t)(uintptr_t)&abuf[1][0];

  auto stage = [&](int tap, uint32_t lb) {
    int dy = tap / 3 - 1, dx = tap % 3 - 1;
    int iy = y_s + dy, ix = x_s + dx;
    int sy = UPSAMPLE ? (iy >> 1) : iy;
    int sx = UPSAMPLE ? (ix >> 1) : ix;
    sy = sy < 0 ? 0 : (sy > Hi - 1 ? Hi - 1 : sy);   // clamp -> legal address
    sx = sx < 0 ? 0 : (sx > Wi - 1 ? Wi - 1 : sx);
    uint64_t g = inB + ((uint64_t)(b_s * Hi + sy) * Wi + sx) * (CIN * 4) +
                 (uint32_t)(seg * (CIN / 2));
    uint32_t l = lb + (uint32_t)(r_s * (AROW * 4) + seg * (CIN / 2));
#pragma unroll
    for (int j = 0; j < CIN / 32; ++j) {
      uint32_t lj = l + j * 16;
      uint64_t gj = g + (uint32_t)(j * 16);
      asm volatile("global_load_async_to_lds_b128 %0, %1, off"
                   :: "v"(lj), "v"(gj) : "memory");
    }
  };

  stage(0, lb0);
  asm volatile("s_wait_asynccnt 0x0" ::: "memory");
  __syncthreads();

  v8f acc = {};
  for (int tap = 0; tap < 9; ++tap) {
    const int cur = tap & 1;
    if (tap < 8) stage(tap + 1, cur ? lb0 : lb1);

    int dy = tap / 3 - 1, dx = tap % 3 - 1;
    int iy = y_l + dy, ix = x_l + dx;
    bool valid = (iy >= 0) & (iy < Ho) & (ix >= 0) & (ix < Wo);
    const float* abase = &abuf[cur][lr * AROW + 2 * half];
    const float* bbase =
        Bw + ((size_t)(tap * (CIN / 4)) * Nout + n0 + lr) * 4 + 2 * half;
#pragma unroll
    for (int g = 0; g < CIN / 32; ++g) {  // groups of 8 K-chunks
      v2f a[8], bf[8];
#pragma unroll
      for (int j = 0; j < 8; ++j) {
        int cc = g * 8 + j;
        a[j] = *(const v2f*)(abase + cc * 4);
        if (!valid) { a[j].x = 0.f; a[j].y = 0.f; }
        bf[j] = *(const v2f*)(bbase + (size_t)cc * Nout * 4);
      }
#pragma unroll
      for (int j = 0; j < 8; ++j) acc = wmma4(a[j], bf[j], acc);
    }
    if (tap < 8) {
      asm volatile("s_wait_asynccnt 0x0" ::: "memory");  // my stage done
      __syncthreads();                                   // everyone's done
    }
  }

  float bv = bias[n0 + lr];
#pragma unroll
  for (int r = 0; r < 8; ++r) {
    int mm = m0 + r + 8 * half;
    float v = acc[r] + bv;
    if (RELU) v = fmaxf(v, 0.f);
    out[(size_t)mm * Nout + n0 + lr] = v;
  }
}

// ---------------------------------------------------------------------------
// Simple per-wave implicit-GEMM conv (kept for the tiny final layer).
// ---------------------------------------------------------------------------
template <bool UPSAMPLE, bool RELU, bool OUT_NCHW>
__global__ __launch_bounds__(128) void conv3_kernel(
    const float* __restrict__ in, const float* __restrict__ Bw,
    const float* __restrict__ bias, float* __restrict__ out, int Ho, int Wo,
    int Cin, int Nout) {
  const int ntile = Nout >> 4;
  int wid = blockIdx.x * 4 + (threadIdx.x >> 5);
  int mt = wid / ntile, nt = wid % ntile;
  int lane = threadIdx.x & 31, half = lane >> 4, lr = lane & 15;
  int m0 = mt * 16, n0 = nt * 16;

  int hw = Ho * Wo;
  int m = m0 + lr;
  int b = m / hw;
  int rem = m % hw;
  int y = rem / Wo, x = rem % Wo;
  int Hi = UPSAMPLE ? (Ho >> 1) : Ho;
  int Wi = UPSAMPLE ? (Wo >> 1) : Wo;

  const int kch = Cin >> 2;
  const int ngrp = kch >> 3;
  v8f acc = {};
  for (int tap = 0; tap < 9; ++tap) {
    int dy = tap / 3 - 1, dx = tap % 3 - 1;
    int iy = y + dy, ix = x + dx;
    bool valid = (iy >= 0) & (iy < Ho) & (ix >= 0) & (ix < Wo);
    int sy = UPSAMPLE ? (iy >> 1) : iy;
    int sx = UPSAMPLE ? (ix >> 1) : ix;
    sy = sy < 0 ? 0 : (sy > Hi - 1 ? Hi - 1 : sy);
    sx = sx < 0 ? 0 : (sx > Wi - 1 ? Wi - 1 : sx);
    const float* abase =
        in + ((size_t)(b * Hi + sy) * Wi + sx) * Cin + 2 * half;
    const float* bbase =
        Bw + ((size_t)(tap * kch) * Nout + n0 + lr) * 4 + 2 * half;
    for (int g = 0; g < ngrp; ++g) {
      v2f a[8], bf[8];
#pragma unroll
      for (int j = 0; j < 8; ++j) {
        int cc = g * 8 + j;
        a[j] = *(const v2f*)(abase + cc * 4);
        if (!valid) { a[j].x = 0.f; a[j].y = 0.f; }
        bf[j] = *(const v2f*)(bbase + (size_t)cc * Nout * 4);
      }
#pragma unroll
      for (int j = 0; j < 8; ++j) acc = wmma4(a[j], bf[j], acc);
    }
  }
  float bv = bias[n0 + lr];
#pragma unroll
  for (int r = 0; r < 8; ++r) {
    int mm = m0 + r + 8 * half;
    float v = acc[r] + bv;
    if (RELU) v = fmaxf(v, 0.f);
    if (OUT_NCHW) {
      int n = n0 + lr;
      if (n < 3) {
        int bb = mm / hw, rr = mm % hw;
        out[(size_t)(bb * 3 + n) * hw + rr] = v;
      }
    } else {
      out[(size_t)mm * Nout + n0 + lr] = v;
    }
  }
}

// ---------------------------------------------------------------------------
extern "C" void kernel_launch(void* const* d_in, const int* in_sizes, int n_in,
                              void* d_out, int out_size, void* d_ws,
                              size_t ws_size, hipStream_t stream) {
  (void)in_sizes; (void)n_in; (void)out_size; (void)ws_size;
  const float* z     = (const float*)d_in[0];   // (4,256,32,32)
  const float* embed = (const float*)d_in[1];   // (16384,256)
  const float* pq_w  = (const float*)d_in[2];   // (256,256,1,1)
  const float* pq_b  = (const float*)d_in[3];
  const float* w_in  = (const float*)d_in[4];   // (128,256,3,3)
  const float* b_in  = (const float*)d_in[5];
  const float* w_u1  = (const float*)d_in[6];   // (128,128,3,3)
  const float* b_u1  = (const float*)d_in[7];
  const float* w_u2  = (const float*)d_in[8];   // (64,128,3,3)
  const float* b_u2  = (const float*)d_in[9];
  const float* w_out = (const float*)d_in[10];  // (3,64,3,3)
  const float* b_out = (const float*)d_in[11];
  float* out = (float*)d_out;                   // (4,3,128,128)

  char* ws = (char*)d_ws;
  size_t off = 0;
  auto alloc = [&](size_t bytes) -> void* {
    void* p = ws + off;
    off = (off + bytes + 255) & ~(size_t)255;
    return p;
  };
  int*   idx   = (int*)  alloc(4096 * 4);
  float* enorm = (float*)alloc((size_t)NEMB * 4);
  float* Bpq   = (float*)alloc((size_t)256 * 256 * 4);
  float* Bin   = (float*)alloc((size_t)2304 * 128 * 4);
  float* Bu1   = (float*)alloc((size_t)1152 * 128 * 4);
  float* Bu2   = (float*)alloc((size_t)1152 * 64 * 4);
  float* Bout  = (float*)alloc((size_t)576 * 16 * 4);
  float* bpad  = (float*)alloc(16 * 4);
  float* q     = (float*)alloc((size_t)4096 * 256 * 4);   // NHWC 32x32x256
  float* h1    = (float*)alloc((size_t)4096 * 128 * 4);   // NHWC 32x32x128
  float* h2    = (float*)alloc((size_t)16384 * 128 * 4);  // NHWC 64x64x128
  float* h3    = (float*)alloc((size_t)65536 * 64 * 4);   // NHWC 128x128x64

  // prep (independent; all on stream, ordered before consumers)
  enorm_kernel<<<NEMB / 256, 256, 0, stream>>>(embed, enorm);
  repack_pq_kernel<<<(256 * 256) / 256, 256, 0, stream>>>(pq_w, Bpq);
  repack_conv3_kernel<<<(9 * 256 * 128) / 256, 256, 0, stream>>>(w_in, Bin, 256, 128);
  repack_conv3_kernel<<<(9 * 128 * 128) / 256, 256, 0, stream>>>(w_u1, Bu1, 128, 128);
  repack_conv3_kernel<<<(9 * 128 * 64) / 256, 256, 0, stream>>>(w_u2, Bu2, 128, 64);
  repack_convout_kernel<<<(9 * 64 * 16) / 256, 256, 0, stream>>>(w_out, b_out, Bout, bpad);

  // VQ argmin: 256 row tiles x 4 waves (async double-buffered LDS pipeline)
  vq_argmin_kernel<<<256, 128, 0, stream>>>(z, embed, enorm, idx);

  // pq 1x1 conv (gather GEMM): 4096 wave-tiles / 4 waves per block
  pq_gemm_kernel<<<1024, 128, 0, stream>>>(embed, idx, Bpq, pq_b, q);

  // conv_in: 32x32, 256->128, relu   (256 M-tiles x 2 N-groups)
  conv3s_kernel<256, false, true><<<512, 128, 0, stream>>>(
      q, Bin, b_in, h1, 32, 32, 128);
  // up2 + conv_u1: 64x64, 128->128, relu  (1024 x 2)
  conv3s_kernel<128, true, true><<<2048, 128, 0, stream>>>(
      h1, Bu1, b_u1, h2, 64, 64, 128);
  // up2 + conv_u2: 128x128, 128->64, relu (4096 x 1)
  conv3s_kernel<128, true, true><<<4096, 128, 0, stream>>>(
      h2, Bu2, b_u2, h3, 128, 128, 64);
  // conv_out: 128x128, 64->3 (N padded to 16), NCHW store (4096 wave-tiles)
  conv3_kernel<false, false, true><<<1024, 128, 0, stream>>>(
      h3, Bout, bpad, out, 128, 128, 64, 16);
}